// ChebyNet_60601988547227
// MI455X (gfx1250) — compile-verified
//
#include <hip/hip_runtime.h>
#include <math.h>

// ---------------------------------------------------------------------------
// ChebyNet (K=3, F=64, UNITS=64) for MI455X / gfx1250.
//   stage 1: edge normalization (atomic degree + rsqrt + per-edge w_hat)
//   stage 2: two sparse propagates (scatter-atomic SpMV, L2-resident)
//   stage 3: fused [T0|T1|T2](Nx192) @ [K0;K1;K2](192x64) via
//            V_WMMA_F32_16X16X4_F32, weights staged pair-interleaved in LDS.
// ---------------------------------------------------------------------------

#define FDIM  64
#define UDIM  64
#define KCHEB 3
#define LAMBDA_MAX 2.0f

typedef __attribute__((ext_vector_type(2))) float v2f;
typedef __attribute__((ext_vector_type(8))) float v8f;

// Hardware f32 atomic add (GLOBAL_ATOMIC_ADD_F32); segment_sum order is
// nondeterministic in the reference too, so the fast path is fine.
__device__ __forceinline__ void atomic_add_f32(float* p, float v) {
    unsafeAtomicAdd(p, v);
}

// ---------------- elementwise helpers ----------------

__global__ void zero_kernel(float* __restrict__ p, int n) {
    int i = blockIdx.x * blockDim.x + threadIdx.x;
    if (i < n) p[i] = 0.0f;
}

// dst[i] = a * src[i]   (seeds SpMV output with the L_hat diagonal term)
__global__ void scale_set_kernel(float* __restrict__ dst, const float* __restrict__ src,
                                 float a, int n) {
    int i = blockIdx.x * blockDim.x + threadIdx.x;
    if (i < n) dst[i] = a * src[i];
}

// t2 = 2*tmp - t0
__global__ void t2_kernel(const float* __restrict__ tmp, const float* __restrict__ t0,
                          float* __restrict__ t2, int n) {
    int i = blockIdx.x * blockDim.x + threadIdx.x;
    if (i < n) t2[i] = 2.0f * tmp[i] - t0[i];
}

// ---------------- edge normalization ----------------

__global__ void deg_kernel(const int* __restrict__ row, const int* __restrict__ col,
                           const float* __restrict__ w, float* __restrict__ deg, int E) {
    int e = blockIdx.x * blockDim.x + threadIdx.x;
    if (e >= E) return;
    int r = row[e];
    float we = (r == col[e]) ? 0.0f : w[e];   // self-loops removed
    if (we != 0.0f) atomic_add_f32(deg + r, we);
}

__global__ void dinv_kernel(const float* __restrict__ deg, float* __restrict__ dinv, int n) {
    int i = blockIdx.x * blockDim.x + threadIdx.x;
    if (i < n) {
        float d = deg[i];
        dinv[i] = (d > 0.0f) ? rsqrtf(d) : 0.0f;
    }
}

__global__ void what_kernel(const int* __restrict__ row, const int* __restrict__ col,
                            const float* __restrict__ w, const float* __restrict__ dinv,
                            float* __restrict__ w_hat, float scale, int E) {
    int e = blockIdx.x * blockDim.x + threadIdx.x;
    if (e >= E) return;
    int r = row[e], c = col[e];
    float we = (r == c) ? 0.0f : w[e];
    w_hat[e] = -dinv[r] * we * dinv[c] * scale;
}

// ---------------- sparse propagate: dst[row] += w_hat * src[col] ----------------
// 16 threads per edge, 4 contiguous floats each -> 256B coalesced gather/scatter.
// x / T1 / dst are 12.8MB each: fully resident in the 192MB L2, so this stage is
// bounded by L2 f32-atomic throughput, not HBM.

__global__ void spmv_scatter(const int* __restrict__ row, const int* __restrict__ col,
                             const float* __restrict__ w_hat,
                             const float* __restrict__ src, float* __restrict__ dst, int E) {
    int t = blockIdx.x * blockDim.x + threadIdx.x;
    int e = t >> 4;
    if (e >= E) return;
    float w = w_hat[e];
    if (w == 0.0f) return;                 // zero-weight / self-loop edges contribute nothing
    int fo = (t & 15) * 4;
    const float* s = src + (size_t)col[e] * FDIM + fo;
    float*       d = dst + (size_t)row[e] * FDIM + fo;
    atomic_add_f32(d + 0, w * s[0]);
    atomic_add_f32(d + 1, w * s[1]);
    atomic_add_f32(d + 2, w * s[2]);
    atomic_add_f32(d + 3, w * s[3]);
}

// ---------------- fused Chebyshev GEMM via V_WMMA_F32_16X16X4_F32 ----------------
// out(Nx64) = [T0|T1|T2](Nx192) @ W(192x64) + bias.  W = kernels (K,F,U) contiguous.
// One wave32 computes one 16-row x 64-col tile: 4 accumulators, 48 K-steps of 4.
//
// LDS weight image is pair-interleaved: sW[kpair][n][j] = W[2*kpair+j][n], so each
// B-fragment (two K-adjacent floats for one column) is a single aligned ds_load_b64
// straight into an even VGPR pair — no register shuffles. Each k-pair row is padded
// 128->160 floats: stride mod 64 banks = 32, so the two lane-halves (k-pair p vs
// p+1) hit disjoint halves of the 64 LDS banks -> conflict-free b64 reads.
//
// Operand layouts (ISA 7.12.2, wave32):
//   A (16x4 f32, 2 VGPRs): lanes 0-15 -> M=lane, {K0,K1}; lanes 16-31 -> M=lane-16, {K2,K3}
//   B (4x16 f32, 2 VGPRs): lanes 0-15 -> N=lane, {K0,K1}; lanes 16-31 -> N=lane-16, {K2,K3}
//   C/D (16x16 f32, 8 VGPRs): vgpr v, lanes 0-15 -> M=v, N=lane; lanes 16-31 -> M=v+8

#define NPAIRS      (KCHEB * FDIM / 2)   // 96 K-row pairs
#define PAIR_STRIDE (UDIM * 2 + 32)      // 160 floats (bank-rotating pad)

__global__ void __launch_bounds__(256)
cheb_gemm_wmma(const float* __restrict__ t0, const float* __restrict__ t1,
               const float* __restrict__ t2, const float* __restrict__ W,
               const float* __restrict__ bias, float* __restrict__ out, int n_nodes) {
    __shared__ float sW[NPAIRS * PAIR_STRIDE];           // 60KB in LDS
    for (int i = threadIdx.x; i < KCHEB * FDIM * UDIM; i += blockDim.x) {
        int k = i / UDIM, n = i - k * UDIM;
        sW[(k >> 1) * PAIR_STRIDE + n * 2 + (k & 1)] = W[i];
    }
    __syncthreads();

    const int lane = threadIdx.x & 31;
    const int wave = threadIdx.x >> 5;
    const int tile = blockIdx.x * (blockDim.x >> 5) + wave;   // 16-row tile (wave-uniform)
    const int m0   = tile * 16;
    if (m0 >= n_nodes) return;                                // uniform exit: EXEC stays all-1s

    const int half = lane >> 4;     // 0: K pair {0,1}, 1: K pair {2,3}
    const int l    = lane & 15;

    v8f acc0 = {}, acc1 = {}, acc2 = {}, acc3 = {};

#pragma unroll
    for (int kk = 0; kk < KCHEB; ++kk) {
        const float* T = (kk == 0) ? t0 : ((kk == 1) ? t1 : t2);   // compile-time select
        const float* arow  = T + (size_t)(m0 + l) * FDIM + half * 2;
        const float* bbase = sW + (kk * (FDIM / 2) + half) * PAIR_STRIDE + 2 * l;
        for (int s = 0; s < FDIM / 4; ++s) {
            v2f a;
            a.x = arow[4 * s + 0];
            a.y = arow[4 * s + 1];

            const float* bp = bbase + 2 * s * PAIR_STRIDE;   // k-pair = kk*32 + 2s + half
            v2f b0 = *(const v2f*)(bp + 0);                  // cols  0..15 of this N-tile set
            v2f b1 = *(const v2f*)(bp + 32);
            v2f b2 = *(const v2f*)(bp + 64);
            v2f b3 = *(const v2f*)(bp + 96);
            acc0 = __builtin_amdgcn_wmma_f32_16x16x4_f32(false, a, false, b0, (short)0, acc0, false, false);
            acc1 = __builtin_amdgcn_wmma_f32_16x16x4_f32(false, a, false, b1, (short)0, acc1, false, false);
            acc2 = __builtin_amdgcn_wmma_f32_16x16x4_f32(false, a, false, b2, (short)0, acc2, false, false);
            acc3 = __builtin_amdgcn_wmma_f32_16x16x4_f32(false, a, false, b3, (short)0, acc3, false, false);
        }
    }

    const float bia0 = bias[l], bia1 = bias[16 + l], bia2 = bias[32 + l], bia3 = bias[48 + l];
#pragma unroll
    for (int v = 0; v < 8; ++v) {
        const int rowi = m0 + v + half * 8;
        float* o = out + (size_t)rowi * UDIM;
        o[l]      = acc0[v] + bia0;
        o[16 + l] = acc1[v] + bia1;
        o[32 + l] = acc2[v] + bia2;
        o[48 + l] = acc3[v] + bia3;
    }
}

// ---------------- host-side launch ----------------

extern "C" void kernel_launch(void* const* d_in, const int* in_sizes, int n_in,
                              void* d_out, int out_size, void* d_ws, size_t ws_size,
                              hipStream_t stream) {
    const float* x       = (const float*)d_in[0];        // N x 64
    const int*   eidx    = (const int*)d_in[1];          // 2 x E
    const float* ew      = (const float*)d_in[2];        // E
    const float* kernels = (const float*)d_in[3];        // 3 x 64 x 64
    const float* bias    = (const float*)d_in[4];        // 64
    float*       out     = (float*)d_out;                // N x 64

    const int N = in_sizes[0] / FDIM;
    const int E = in_sizes[2];
    const int* row = eidx;
    const int* col = eidx + E;

    // workspace partition (floats): deg[N] dinv[N] w_hat[E] T1[N*64] TMP[N*64] T2[N*64]
    float* ws    = (float*)d_ws;
    float* deg   = ws;
    float* dinv  = deg  + N;
    float* w_hat = dinv + N;
    float* T1    = w_hat + E;
    float* TMP   = T1 + (size_t)N * FDIM;
    float* T2    = TMP + (size_t)N * FDIM;

    const float scale = 2.0f / LAMBDA_MAX;    // = 1.0
    const float diag  = scale - 1.0f;         // = 0.0  (L_hat diagonal)
    const int NF = N * FDIM;
    const int TB = 256;

    // stage 1: edge normalization
    zero_kernel <<<(N + TB - 1) / TB, TB, 0, stream>>>(deg, N);
    deg_kernel  <<<(E + TB - 1) / TB, TB, 0, stream>>>(row, col, ew, deg, E);
    dinv_kernel <<<(N + TB - 1) / TB, TB, 0, stream>>>(deg, dinv, N);
    what_kernel <<<(E + TB - 1) / TB, TB, 0, stream>>>(row, col, ew, dinv, w_hat, scale, E);

    // stage 2: Chebyshev recurrence
    const int spmv_threads = E * 16;
    scale_set_kernel<<<(NF + TB - 1) / TB, TB, 0, stream>>>(T1, x, diag, NF);
    spmv_scatter    <<<(spmv_threads + TB - 1) / TB, TB, 0, stream>>>(row, col, w_hat, x, T1, E);
    scale_set_kernel<<<(NF + TB - 1) / TB, TB, 0, stream>>>(TMP, T1, diag, NF);
    spmv_scatter    <<<(spmv_threads + TB - 1) / TB, TB, 0, stream>>>(row, col, w_hat, T1, TMP, E);
    t2_kernel       <<<(NF + TB - 1) / TB, TB, 0, stream>>>(TMP, x, T2, NF);

    // stage 3: fused GEMM + bias (WMMA)
    const int tiles = (N + 15) / 16;
    const int waves_per_block = TB / 32;
    cheb_gemm_wmma<<<(tiles + waves_per_block - 1) / waves_per_block, TB, 0, stream>>>(
        x, T1, T2, kernels, bias, out, N);
}